// FocalLossModified_17102559773189
// MI455X (gfx1250) — compile-verified
//
#include <hip/hip_runtime.h>

typedef __attribute__((ext_vector_type(2))) float v2f;
typedef __attribute__((ext_vector_type(4))) float v4f;
typedef __attribute__((ext_vector_type(8))) float v8f;

// XOR-pattern lane shuffle+add via ds_swizzle_b32 (group-of-32 mode:
// offset[4:0]=and_mask=0x1f, offset[9:5]=or_mask=0, offset[14:10]=xor_mask).
// Equivalent to v += __shfl_xor(v, MASK, 32) when EXEC is all ones.
template <int XOR_MASK>
__device__ __forceinline__ float lane_xor_add(float v) {
  int sw = __builtin_amdgcn_ds_swizzle(__builtin_bit_cast(int, v),
                                       (XOR_MASK << 10) | 0x1f);
  return v + __builtin_bit_cast(float, sw);
}

// Wave32 sum-reduction using V_WMMA_F32_16X16X4_F32 with A = ones(16x4).
// D[m,n] = sum_k B[k,n]; with B-vgpr0 = x, B-vgpr1 = 0 the two half-wave rows
// of vgpr0 are distinct K rows, so every lane's c[0] = x[n] + x[n+16] with
// n = lane&15. 4 XOR-swizzle steps over the 16 distinct values finish the
// full 32-lane sum. Requires EXEC == all ones (fully converged code only).
__device__ __forceinline__ float wave_sum_wmma(float x) {
  v2f a; a.x = 1.0f; a.y = 1.0f;      // A matrix: all ones
  v2f b; b.x = x;    b.y = 0.0f;      // B matrix: data in vgpr0, zeros in vgpr1
  v8f c = {0.f, 0.f, 0.f, 0.f, 0.f, 0.f, 0.f, 0.f};
  c = __builtin_amdgcn_wmma_f32_16x16x4_f32(false, a, false, b, (short)0, c,
                                            false, false);
  float v = c[0];
  v = lane_xor_add<8>(v);
  v = lane_xor_add<4>(v);
  v = lane_xor_add<2>(v);
  v = lane_xor_add<1>(v);
  return v;  // full-wave sum, replicated to all lanes
}

#define NCLS 80
#define NGT  32

__global__ void focal_main_kernel(const float* __restrict__ cls,
                                  const float* __restrict__ reg,
                                  const float* __restrict__ anchors,
                                  const float* __restrict__ ann,
                                  float* __restrict__ ws, int A) {
  const int b = blockIdx.y;
  const int i = blockIdx.x * 256 + threadIdx.x;

  __shared__ float s_ann[NGT * 6];
  __shared__ float s_red[8][3];

  if (threadIdx.x < NGT * 6)
    s_ann[threadIdx.x] = ann[(size_t)b * (NGT * 6) + threadIdx.x];
  __syncthreads();

  float cls_part = 0.0f, reg_part = 0.0f, pos_part = 0.0f;

  if (i < A) {
    // anchors are reused by all 8 images -> default (RT) loads, stay in L2
    const float ax1 = anchors[(size_t)i * 4 + 0];
    const float ay1 = anchors[(size_t)i * 4 + 1];
    const float ax2 = anchors[(size_t)i * 4 + 2];
    const float ay2 = anchors[(size_t)i * 4 + 3];
    const float aw = ax2 - ax1, ah = ay2 - ay1;
    const float acx = ax1 + 0.5f * aw, acy = ay1 + 0.5f * ah;
    const float areaA = aw * ah;

    // IoU max / argmax (first-max semantics like jnp.argmax)
    float best = -2.0f;
    int bi = 0;
#pragma unroll 8
    for (int m = 0; m < NGT; ++m) {
      const float gx1 = s_ann[m * 6 + 0];
      const float gy1 = s_ann[m * 6 + 1];
      const float gx2 = s_ann[m * 6 + 2];
      const float gy2 = s_ann[m * 6 + 3];
      const float gcl = s_ann[m * 6 + 4];
      float iw = fminf(ax2, gx2) - fmaxf(ax1, gx1);
      float ih = fminf(ay2, gy2) - fmaxf(ay1, gy1);
      float inter = fmaxf(iw, 0.0f) * fmaxf(ih, 0.0f);
      float ua = fmaxf(areaA + (gx2 - gx1) * (gy2 - gy1) - inter, 1e-8f);
      float iou = inter / ua;
      if (gcl == -1.0f) iou = -1.0f;   // padding mask
      if (iou > best) { best = iou; bi = m; }
    }

    const bool pos = (best >= 0.5f);
    const bool neg = (best < 0.4f);
    pos_part = pos ? 1.0f : 0.0f;

    if (pos || neg) {  // ignore band (0.4 <= iou < 0.5) contributes nothing
      const int cid = pos ? (int)s_ann[bi * 6 + 4] : -1;
      const v4f* cp =
          reinterpret_cast<const v4f*>(cls + ((size_t)b * A + i) * NCLS);
      float acc = 0.0f;
#pragma unroll 10
      for (int c4 = 0; c4 < NCLS / 4; ++c4) {
        // 256 MB streamed exactly once -> non-temporal load
        v4f v4 = __builtin_nontemporal_load(cp + c4);
#pragma unroll
        for (int k = 0; k < 4; ++k) {
          const int c = c4 * 4 + k;
          float p = fminf(fmaxf(v4[k], 1e-4f), 1.0f - 1e-4f);
          // target==1: 0.25*(1-p)^2*(-log p); target==0: 0.75*p^2*(-log(1-p))
          float q  = (c == cid) ? p : 1.0f - p;
          float af = (c == cid) ? 0.25f : 0.75f;
          float om = 1.0f - q;
          acc += af * om * om * (-__builtin_logf(q));
        }
      }
      cls_part = acc;
    }

    if (pos) {
      const float gx1 = s_ann[bi * 6 + 0];
      const float gy1 = s_ann[bi * 6 + 1];
      const float gx2 = s_ann[bi * 6 + 2];
      const float gy2 = s_ann[bi * 6 + 3];
      float gwu = gx2 - gx1, ghu = gy2 - gy1;
      float gcx = gx1 + 0.5f * gwu, gcy = gy1 + 0.5f * ghu;
      float gw = fmaxf(gwu, 1.0f), gh = fmaxf(ghu, 1.0f);
      float t0 = ((gcx - acx) / aw) / 0.1f;
      float t1 = ((gcy - acy) / ah) / 0.1f;
      float t2 = __builtin_logf(gw / aw) / 0.2f;
      float t3 = __builtin_logf(gh / ah) / 0.2f;
      v4f r4 = __builtin_nontemporal_load(
          reinterpret_cast<const v4f*>(reg + ((size_t)b * A + i) * 4));
      float tt[4] = {t0, t1, t2, t3};
      float racc = 0.0f;
#pragma unroll
      for (int k = 0; k < 4; ++k) {
        float d = fabsf(tt[k] - r4[k]);
        racc += (d <= (1.0f / 9.0f)) ? 4.5f * d * d : d - (0.5f / 9.0f);
      }
      reg_part = racc;
    }
  }

  // ---- block reduction: WMMA wave reduce -> LDS -> thread 0 (fixed order) ----
  const float wc = wave_sum_wmma(cls_part);
  const float wr = wave_sum_wmma(reg_part);
  const float wp = wave_sum_wmma(pos_part);
  const int wave = threadIdx.x >> 5;
  if ((threadIdx.x & 31) == 0) {
    s_red[wave][0] = wc;
    s_red[wave][1] = wr;
    s_red[wave][2] = wp;
  }
  __syncthreads();
  if (threadIdx.x == 0) {
    float tc = 0.0f, tr = 0.0f, tp = 0.0f;
#pragma unroll
    for (int w = 0; w < 8; ++w) {
      tc += s_red[w][0];
      tr += s_red[w][1];
      tp += s_red[w][2];
    }
    const size_t slot = ((size_t)b * gridDim.x + blockIdx.x) * 3;
    ws[slot + 0] = tc;
    ws[slot + 1] = tr;
    ws[slot + 2] = tp;
  }
}

__global__ void focal_finalize_kernel(const float* __restrict__ ws,
                                      float* __restrict__ out, int nbx, int B) {
  __shared__ float s_cls[32];
  __shared__ float s_reg[32];
  const int wave = threadIdx.x >> 5;
  const int lane = threadIdx.x & 31;
  // grid is 32*B threads: every launched wave has wave < B, EXEC stays full
  float c = 0.0f, r = 0.0f, p = 0.0f;
  for (int j = lane; j < nbx; j += 32) {
    const size_t slot = ((size_t)wave * nbx + j) * 3;
    c += ws[slot + 0];
    r += ws[slot + 1];
    p += ws[slot + 2];
  }
  c = lane_xor_add<16>(c); r = lane_xor_add<16>(r); p = lane_xor_add<16>(p);
  c = lane_xor_add<8>(c);  r = lane_xor_add<8>(r);  p = lane_xor_add<8>(p);
  c = lane_xor_add<4>(c);  r = lane_xor_add<4>(r);  p = lane_xor_add<4>(p);
  c = lane_xor_add<2>(c);  r = lane_xor_add<2>(r);  p = lane_xor_add<2>(p);
  c = lane_xor_add<1>(c);  r = lane_xor_add<1>(r);  p = lane_xor_add<1>(p);
  if (lane == 0) {
    const float denom = fmaxf(p, 1.0f);
    s_cls[wave] = c / denom;
    s_reg[wave] = (p > 0.0f) ? r / (4.0f * denom) : 0.0f;
  }
  __syncthreads();
  if (threadIdx.x == 0) {
    float mc = 0.0f, mr = 0.0f;
    for (int b = 0; b < B; ++b) {
      mc += s_cls[b];
      mr += s_reg[b];
    }
    out[0] = mc / (float)B;
    out[1] = mr / (float)B;
  }
}

extern "C" void kernel_launch(void* const* d_in, const int* in_sizes, int n_in,
                              void* d_out, int out_size, void* d_ws,
                              size_t ws_size, hipStream_t stream) {
  const float* cls = (const float*)d_in[0];  // [B,A,80]
  const float* reg = (const float*)d_in[1];  // [B,A,4]
  const float* anc = (const float*)d_in[2];  // [1,A,4]
  const float* ann = (const float*)d_in[3];  // [B,32,6]
  float* out = (float*)d_out;                // [2]
  float* W = (float*)d_ws;

  const int A = in_sizes[2] / 4;
  const int B = in_sizes[1] / (A * 4);
  const int nbx = (A + 255) / 256;

  dim3 grid(nbx, B);
  focal_main_kernel<<<grid, 256, 0, stream>>>(cls, reg, anc, ann, W, A);
  focal_finalize_kernel<<<1, 32 * B, 0, stream>>>(W, out, nbx, B);
}